// Decoder_87883620811423
// MI455X (gfx1250) — compile-verified
//
#include <hip/hip_runtime.h>
#include <hip/hip_bf16.h>

typedef __attribute__((ext_vector_type(2))) float v2f;
typedef __attribute__((ext_vector_type(8))) float v8f;

#define EMB 128
#define KSTEPS (EMB / 4)          // 32 WMMA f32 16x16x4 steps cover K=128
#define EDGE_BLOCKS 1024
#define EDGE_THREADS 256

// Select element (kk, n) of the padded 128x16 weight matrix built from W (256x2):
// col0 = W[0:128,0], col1 = W[0:128,1], col2 = W[128:256,0], col3 = W[128:256,1], cols 4..15 = 0.
__device__ __forceinline__ float wsel(const float* __restrict__ W, int kk, int n) {
    if (n == 0) return W[kk * 2 + 0];
    if (n == 1) return W[kk * 2 + 1];
    if (n == 2) return W[(EMB + kk) * 2 + 0];
    if (n == 3) return W[(EMB + kk) * 2 + 1];
    return 0.0f;
}

// Kernel 1: P[n][0..3] = Z[n] @ [Wtop0 Wtop1 Wbot0 Wbot1]  via V_WMMA_F32_16X16X4_F32.
// One wave per 16-node tile, grid-stride over tiles. B matrix (constant) is hoisted
// into registers outside the node loop; each step needs only one float2 A-load/lane.
__global__ void __launch_bounds__(256) project_kernel(const float* __restrict__ Z,
                                                      const float* __restrict__ W,
                                                      float* __restrict__ P,
                                                      int n_nodes) {
    const int lane   = threadIdx.x & 31;
    const int wave   = (blockIdx.x * blockDim.x + threadIdx.x) >> 5;
    const int nwaves = (gridDim.x * blockDim.x) >> 5;
    const int n      = lane & 15;            // N column this lane serves in B/C
    const int khalf  = (lane >> 4) * 2;      // K sub-offset: lanes 0-15 -> K+0/1, 16-31 -> K+2/3

    // --- Build B (4x16 per step) for all 32 K-steps, held in VGPRs for the whole loop ---
    v2f barr[KSTEPS];
#pragma unroll
    for (int s = 0; s < KSTEPS; ++s) {
        const int kk = s * 4 + khalf;
        v2f bb;
        bb.x = wsel(W, kk + 0, n);
        bb.y = wsel(W, kk + 1, n);
        barr[s] = bb;
    }

    const int ntiles = (n_nodes + 15) >> 4;
    for (int tile = wave; tile < ntiles; tile += nwaves) {   // wave-uniform: EXEC all-1s at WMMA
        int row = tile * 16 + n;
        if (row >= n_nodes) row = n_nodes - 1;               // per-lane clamp (safe dup load)
        const float* zrow = Z + (size_t)row * EMB + khalf;

        v8f c = {};
#pragma unroll
        for (int s = 0; s < KSTEPS; ++s) {
            v2f a = *(const v2f*)(zrow + s * 4);             // A: (M=row, K=s*4+khalf .. +1)
            // D = A(16x4,f32) x B(4x16,f32) + C(16x16,f32)
            c = __builtin_amdgcn_wmma_f32_16x16x4_f32(
                    /*neg_a=*/false, a, /*neg_b=*/false, barr[s],
                    /*c_mod=*/(short)0, c, /*reuse_a=*/false, /*reuse_b=*/false);
        }

        // C layout: VGPR v, lanes 0-15 -> (M=v, N=lane); lanes 16-31 -> (M=v+8, N=lane-16).
        // Only columns N=0..3 are meaningful.
        if (n < 4) {
            const int baseM = tile * 16 + (lane >> 4) * 8;
#pragma unroll
            for (int v = 0; v < 8; ++v) {
                const int m = baseM + v;
                if (m < n_nodes) P[(size_t)m * 4 + n] = c[v];
            }
        }
    }
}

// Kernel 2: per-edge 2-class NLL using the tiny P table (L2-resident gathers),
// wave32 shuffle reduction + per-block partial sums into workspace.
__global__ void __launch_bounds__(EDGE_THREADS) edge_loss_kernel(
        const long long* __restrict__ edges,   // [n_edges][2] int64
        const long long* __restrict__ y,       // [n_edges] int64
        const float* __restrict__ P,           // [n_nodes][4]
        float* __restrict__ partial,
        int n_edges) {
    float acc = 0.0f;
    const int tid    = blockIdx.x * blockDim.x + threadIdx.x;
    const int stride = gridDim.x * blockDim.x;

    for (int e = tid; e < n_edges; e += stride) {
        const int src = (int)edges[2 * e + 0];
        const int dst = (int)edges[2 * e + 1];
        const float4 ps = ((const float4*)P)[src];
        const float4 pd = ((const float4*)P)[dst];
        const float s0 = ps.x + pd.z;
        const float s1 = ps.y + pd.w;
        const float m  = fmaxf(s0, s1);
        const float lse = m + logf(expf(s0 - m) + expf(s1 - m));
        const float sy  = (y[e] == 0) ? s0 : s1;
        acc += (lse - sy);                      // = -log_softmax(scores)[y]
    }

    // wave32 reduction
#pragma unroll
    for (int off = 16; off > 0; off >>= 1) acc += __shfl_xor(acc, off, 32);

    __shared__ float smem[EDGE_THREADS / 32];
    if ((threadIdx.x & 31) == 0) smem[threadIdx.x >> 5] = acc;
    __syncthreads();
    if (threadIdx.x == 0) {
        float t = 0.0f;
#pragma unroll
        for (int i = 0; i < EDGE_THREADS / 32; ++i) t += smem[i];
        partial[blockIdx.x] = t;
    }
}

// Kernel 3: deterministic single-block reduction of the block partials.
__global__ void __launch_bounds__(1024) final_reduce_kernel(const float* __restrict__ partial,
                                                            float* __restrict__ out,
                                                            int n, float invN) {
    const int t = threadIdx.x;
    float v = (t < n) ? partial[t] : 0.0f;
#pragma unroll
    for (int off = 16; off > 0; off >>= 1) v += __shfl_xor(v, off, 32);

    __shared__ float smem[32];
    if ((t & 31) == 0) smem[t >> 5] = v;
    __syncthreads();
    if (t < 32) {
        float w = (t < (int)(blockDim.x >> 5)) ? smem[t] : 0.0f;
#pragma unroll
        for (int off = 16; off > 0; off >>= 1) w += __shfl_xor(w, off, 32);
        if (t == 0) out[0] = w * invN;
    }
}

extern "C" void kernel_launch(void* const* d_in, const int* in_sizes, int n_in,
                              void* d_out, int out_size, void* d_ws, size_t ws_size,
                              hipStream_t stream) {
    const float*     Z     = (const float*)d_in[0];      // [n_nodes][128] f32
    const long long* edges = (const long long*)d_in[1];  // [n_edges][2]  i64
    const long long* y     = (const long long*)d_in[2];  // [n_edges]     i64
    const float*     W     = (const float*)d_in[3];      // [256][2]      f32

    const int n_nodes = in_sizes[0] / EMB;
    const int n_edges = in_sizes[1] / 2;

    // Workspace layout: P table, then block partials.
    float* P       = (float*)d_ws;                           // n_nodes*4 floats
    float* partial = P + (size_t)n_nodes * 4;                // EDGE_BLOCKS floats

    // Kernel 1: one wave per 16-node tile; 8 waves per 256-thread block.
    const int ntiles = (n_nodes + 15) / 16;
    const int grid1  = (ntiles + 7) / 8;
    project_kernel<<<grid1, 256, 0, stream>>>(Z, W, P, n_nodes);

    // Kernel 2: edge loss + partial sums.
    edge_loss_kernel<<<EDGE_BLOCKS, EDGE_THREADS, 0, stream>>>(edges, y, P, partial, n_edges);

    // Kernel 3: final deterministic reduction; writes d_out unconditionally.
    final_reduce_kernel<<<1, 1024, 0, stream>>>(partial, (float*)d_out, EDGE_BLOCKS,
                                                1.0f / (float)n_edges);
}